// RGCN_40132174414144
// MI455X (gfx1250) — compile-verified
//
#include <hip/hip_runtime.h>
#include <hip/hip_bf16.h>

// ---------------- problem constants (match reference) ----------------
#define N_SRC0 200000
#define N_DST0 50000
#define N_DST1 10000
#define E0     800000
#define E1     160000
#define DIM    256      // IN == HID == OUT
#define NB     4
#define BLK    64       // DIM / NB
#define NPAIR  5000

typedef float v2f __attribute__((ext_vector_type(2)));
typedef float v8f __attribute__((ext_vector_type(8)));

// D = A(16x4 f32) * B(4x16 f32) + C(16x16 f32), exact f32 via CDNA5 WMMA.
__device__ __forceinline__ v8f wmma_f32(v2f a, v2f b, v8f c) {
    return __builtin_amdgcn_wmma_f32_16x16x4_f32(
        /*neg_a=*/false, a, /*neg_b=*/false, b,
        /*c_mod=*/(short)0, c, /*reuse_a=*/false, /*reuse_b=*/false);
}

// ---------------------------------------------------------------------
// Kernel 1: block-diagonal transform  t[n,256] = blockdiag(h[n,256], W[4,64,64])
// Grid: n/16 blocks, 128 threads (4 waves). Wave w handles base-block w:
// t[m0:m0+16, w*64:(w+1)*64] = h[m0:m0+16, w*64:+64] @ W[w]  (16x64x64 GEMM)
// ---------------------------------------------------------------------
__global__ void bdd_transform(const float* __restrict__ h,
                              const float* __restrict__ W,
                              float* __restrict__ t) {
    const int m0   = blockIdx.x * 16;
    const int wv   = threadIdx.x >> 5;   // base block 0..3
    const int lane = threadIdx.x & 31;
    const int row  = lane & 15;          // A row / B,C col within tile
    const int kk   = (lane >> 4) * 2;    // K sub-offset: 0 or 2

    const float* Wb = W + (size_t)wv * BLK * BLK;          // 64x64 row-major
    const float* A  = h + (size_t)m0 * DIM + wv * BLK;     // lda = DIM

    v8f acc[4] = {};                                       // 4 N-tiles of 16
    for (int k = 0; k < BLK; k += 4) {
        const float* ap = A + (size_t)row * DIM + k + kk;
        v2f a = { ap[0], ap[1] };
        #pragma unroll
        for (int nt = 0; nt < 4; ++nt) {
            const float* bp = Wb + (size_t)(k + kk) * BLK + nt * 16 + row;
            v2f b = { bp[0], bp[BLK] };
            acc[nt] = wmma_f32(a, b, acc[nt]);
        }
    }
    float* outp = t + (size_t)m0 * DIM + wv * BLK;
    const int rbase = (lane >> 4) * 8;
    #pragma unroll
    for (int nt = 0; nt < 4; ++nt)
        #pragma unroll
        for (int q = 0; q < 8; ++q)
            outp[(size_t)(rbase + q) * DIM + nt * 16 + row] = acc[nt][q];
}

// ---------------------------------------------------------------------
// Kernel 2: per-edge scatter-add. One wave per edge, 8 f32 atomics/lane.
// agg (<=51MB) stays L2-resident -> cheap L2-side RMW.
// ---------------------------------------------------------------------
__global__ void edge_scatter(const float* __restrict__ t,
                             const int* __restrict__ src,
                             const int* __restrict__ dst,
                             float* __restrict__ agg, int nedges) {
    const int wave = (int)((blockIdx.x * blockDim.x + threadIdx.x) >> 5);
    const int lane = threadIdx.x & 31;
    if (wave >= nedges) return;
    const int s = src[wave], d = dst[wave];
    const float* ts = t + (size_t)s * DIM;
    float*       ad = agg + (size_t)d * DIM;
    #pragma unroll
    for (int i = 0; i < 8; ++i)
        atomicAdd(&ad[lane + 32 * i], ts[lane + 32 * i]);
}

// ---------------------------------------------------------------------
// Kernel 3: out = relu(agg + hprev[:num_dst] @ loopw + b)
// GEMM M=num_dst, N=256, K=256. Grid: num_dst/16 blocks, 128 thr (4 waves);
// wave w covers cols [w*64, w*64+64).
// ---------------------------------------------------------------------
__global__ void loop_fuse(const float* __restrict__ hprev,
                          const float* __restrict__ loopw,
                          const float* __restrict__ bias,
                          const float* __restrict__ agg,
                          float* __restrict__ out) {
    const int m0   = blockIdx.x * 16;
    const int wv   = threadIdx.x >> 5;
    const int lane = threadIdx.x & 31;
    const int col  = lane & 15;
    const int kk   = (lane >> 4) * 2;

    const float* A = hprev + (size_t)m0 * DIM;
    v8f acc[4] = {};
    for (int k = 0; k < DIM; k += 4) {
        const float* ap = A + (size_t)col * DIM + k + kk;
        v2f a = { ap[0], ap[1] };
        #pragma unroll
        for (int nt = 0; nt < 4; ++nt) {
            const int n0 = wv * 64 + nt * 16;
            const float* bp = loopw + (size_t)(k + kk) * DIM + n0 + col;
            v2f b = { bp[0], bp[DIM] };
            acc[nt] = wmma_f32(a, b, acc[nt]);
        }
    }
    const int rbase = (lane >> 4) * 8;
    #pragma unroll
    for (int nt = 0; nt < 4; ++nt) {
        const int n0 = wv * 64 + nt * 16;
        const float bv = bias[n0 + col];
        #pragma unroll
        for (int q = 0; q < 8; ++q) {
            const size_t idx = (size_t)(m0 + rbase + q) * DIM + n0 + col;
            float v = acc[nt][q] + agg[idx] + bv;
            out[idx] = v > 0.f ? v : 0.f;
        }
    }
}

// ---------------------------------------------------------------------
// Kernel 4: link-pred head. e = h[s]*h[d]; hid = relu(e@pw1+pb1) (128);
// score = hid@pw2 + pb2. A-fragment is the gathered product built on the fly.
// Grid: (ceil(NPAIR/16), 2[pos/neg]) x 128 thr. NPAIR not /16 -> clamp+guard.
// ---------------------------------------------------------------------
__global__ void pred_kernel(const float* __restrict__ h,
                            const int* __restrict__ psrc, const int* __restrict__ pdst,
                            const int* __restrict__ nsrc, const int* __restrict__ ndst,
                            const float* __restrict__ pw1, const float* __restrict__ pb1,
                            const float* __restrict__ pw2, const float* __restrict__ pb2,
                            float* __restrict__ out, int npair) {
    __shared__ float hid[16][132];      // 16 pairs x 128 hidden (+pad)

    const int which = blockIdx.y;
    const int* srcA = which ? nsrc : psrc;
    const int* dstA = which ? ndst : pdst;
    float* outp = out + (size_t)which * npair;

    const int m0   = blockIdx.x * 16;
    const int wv   = threadIdx.x >> 5;   // wave covers cols [wv*32, wv*32+32)
    const int lane = threadIdx.x & 31;
    const int row  = lane & 15;
    const int kk   = (lane >> 4) * 2;

    const int m  = m0 + row;
    const int mi = (m < npair) ? m : (npair - 1);       // clamp: EXEC stays full
    const int s = srcA[mi], d = dstA[mi];
    const float* hs = h + (size_t)s * DIM;
    const float* hd = h + (size_t)d * DIM;

    v8f acc[2] = {};
    for (int k = 0; k < DIM; k += 4) {
        v2f a = { hs[k + kk] * hd[k + kk], hs[k + kk + 1] * hd[k + kk + 1] };
        #pragma unroll
        for (int nt = 0; nt < 2; ++nt) {
            const int n0 = wv * 32 + nt * 16;
            const float* bp = pw1 + (size_t)(k + kk) * 128 + n0 + row;
            v2f b = { bp[0], bp[128] };
            acc[nt] = wmma_f32(a, b, acc[nt]);
        }
    }
    const int rbase = (lane >> 4) * 8;
    #pragma unroll
    for (int nt = 0; nt < 2; ++nt) {
        const int n0 = wv * 32 + nt * 16;
        const float bv = pb1[n0 + row];
        #pragma unroll
        for (int q = 0; q < 8; ++q) {
            float v = acc[nt][q] + bv;
            hid[rbase + q][n0 + row] = v > 0.f ? v : 0.f;
        }
    }
    __syncthreads();
    if (threadIdx.x < 16) {
        const int p  = threadIdx.x;
        const int mm = m0 + p;
        if (mm < npair) {
            float sum = pb2[0];
            #pragma unroll 8
            for (int j = 0; j < 128; ++j) sum += hid[p][j] * pw2[j];
            outp[mm] = sum;
        }
    }
}

// ---------------------------------------------------------------------
extern "C" void kernel_launch(void* const* d_in, const int* in_sizes, int n_in,
                              void* d_out, int out_size, void* d_ws, size_t ws_size,
                              hipStream_t stream) {
    const float* x     = (const float*)d_in[0];
    const int* src0    = (const int*)d_in[1];
    const int* dst0    = (const int*)d_in[2];
    const int* src1    = (const int*)d_in[3];
    const int* dst1    = (const int*)d_in[4];
    const int* pos_src = (const int*)d_in[5];
    const int* pos_dst = (const int*)d_in[6];
    const int* neg_src = (const int*)d_in[7];
    const int* neg_dst = (const int*)d_in[8];
    const float* W0    = (const float*)d_in[9];
    const float* loop0 = (const float*)d_in[10];
    const float* b0    = (const float*)d_in[11];
    const float* W1    = (const float*)d_in[12];
    const float* loop1 = (const float*)d_in[13];
    const float* b1    = (const float*)d_in[14];
    const float* pw1   = (const float*)d_in[15];
    const float* pb1   = (const float*)d_in[16];
    const float* pw2   = (const float*)d_in[17];
    const float* pb2   = (const float*)d_in[18];

    float* out = (float*)d_out;                 // [h_pos 5000 | h_neg 5000 | h 10000x256]
    float* ws  = (float*)d_ws;

    // Workspace layout (floats): t reused L0/L1, agg reused L0/L1.
    float* t   = ws;                                        // 200000*256 = 51.2M
    float* agg = ws + (size_t)N_SRC0 * DIM;                 // 50000*256  = 12.8M
    float* h1  = agg + (size_t)N_DST0 * DIM;                // 50000*256  = 12.8M
    float* h2  = out + 2 * NPAIR;                           // final h -> d_out slot

    // ---------------- layer 0 ----------------
    hipMemsetAsync(agg, 0, (size_t)N_DST0 * DIM * sizeof(float), stream);
    bdd_transform<<<N_SRC0 / 16, 128, 0, stream>>>(x, W0, t);
    edge_scatter<<<(E0 + 7) / 8, 256, 0, stream>>>(t, src0, dst0, agg, E0);
    loop_fuse<<<N_DST0 / 16, 128, 0, stream>>>(x, loop0, b0, agg, h1);

    // ---------------- layer 1 ----------------
    hipMemsetAsync(agg, 0, (size_t)N_DST1 * DIM * sizeof(float), stream);
    bdd_transform<<<N_DST0 / 16, 128, 0, stream>>>(h1, W1, t);
    edge_scatter<<<(E1 + 7) / 8, 256, 0, stream>>>(t, src1, dst1, agg, E1);
    loop_fuse<<<N_DST1 / 16, 128, 0, stream>>>(h1, loop1, b1, agg, h2);

    // ---------------- prediction head ----------------
    pred_kernel<<<dim3((NPAIR + 15) / 16, 2), 128, 0, stream>>>(
        h2, pos_src, pos_dst, neg_src, neg_dst, pw1, pb1, pw2, pb2, out, NPAIR);
}